// REINFORCE_32366873543408
// MI455X (gfx1250) — compile-verified
//
#include <hip/hip_runtime.h>

// ---------------------------------------------------------------------------
// REINFORCE reverse-discounted scan for MI455X (gfx1250).
// Scan parallelized as block-triangular matmul via V_WMMA_F32_16X16X4_F32.
// ---------------------------------------------------------------------------

typedef __attribute__((ext_vector_type(2))) float v2f;
typedef __attribute__((ext_vector_type(8))) float v8f;

constexpr int   Bn    = 1024;
constexpr int   Tn    = 8192;
constexpr float GAMMA = 0.99f;
constexpr float CLIPV = 5.0f;
constexpr float EPSc  = 1e-7f;

constexpr int CHUNK = 256;          // timesteps per wave-chunk
constexpr int NCH   = Tn / CHUNK;   // 32 chunks per row
constexpr int TILES = CHUNK / 16;   // 16 WMMA tiles per chunk
constexpr int WPB   = 2;            // waves per block (kernel 1)
constexpr int LSTR  = 257;          // padded LDS row stride (floats) -> no bank conflicts

__device__ __forceinline__ float gpow(int d) {
  float g = 1.0f;
  for (int i = 0; i < d; ++i) g *= GAMMA;
  return g;
}

__device__ __forceinline__ float clampv(float x) {
  return fminf(fmaxf(x, -CLIPV), CLIPV);
}

// ---------------------------------------------------------------------------
// Kernel 1: per-(16-row strip, 256-step chunk) local reverse discounted scan.
// One wave per unit. A-operand = constant scan matrix Mat[t][s]=gamma^(s-t),
// B-operand = r2 data tile from LDS, D = local suffix sums (f32 all the way).
// Writes local sums into the cum half of d_out; chunk-start prefix into Sbuf.
// ---------------------------------------------------------------------------
__global__ void k_local_scan(const float* __restrict__ logits,
                             const float* __restrict__ weight,
                             float* __restrict__ cum,
                             float* __restrict__ Sbuf)
{
  __shared__ float sm_all[WPB][16 * LSTR];
  const int lane  = threadIdx.x & 31;
  const int wave  = threadIdx.x >> 5;
  const int unit  = blockIdx.x * WPB + wave;   // 0 .. (B/16)*NCH - 1
  const int strip = unit / NCH;
  const int chunk = unit % NCH;
  const int b0    = strip * 16;
  const int t0    = chunk * CHUNK;
  float* sm = sm_all[wave];

  // ---- Stage r2 = w^2 * log(sigmoid(x)+eps) into LDS (coalesced 128B lines)
  for (int i = lane; i < 16 * CHUNK; i += 32) {
    const int r = i >> 8;             // i / CHUNK
    const int c = i & (CHUNK - 1);
    const int g = (b0 + r) * Tn + t0 + c;
    const float w   = weight[g];
    const float x   = logits[g];
    const float sig = 1.0f / (1.0f + __expf(-x));
    sm[r * LSTR + c] = w * w * __logf(sig + EPSc);
  }
  __syncthreads();

  const int ln = lane & 15;   // A: time column t   | B: batch row n
  const int hi = lane >> 4;   // K-half selector per ISA 32-bit A/B layout

  // A fragments: Mat[t][s] = gamma^(s-t) for s>=t, else 0 (compile-time-ish)
  v2f afrag[4];
#pragma unroll
  for (int kk = 0; kk < 4; ++kk) {
    const int s0 = 4 * kk + 2 * hi;
    afrag[kk].x = (s0     >= ln) ? gpow(s0     - ln) : 0.0f;
    afrag[kk].y = (s0 + 1 >= ln) ? gpow(s0 + 1 - ln) : 0.0f;
  }
  // Carry factors: D element in VGPR v has time-in-tile (v + 8*hi)
  float gf[8];
#pragma unroll
  for (int v = 0; v < 8; ++v) gf[v] = gpow(16 - v - 8 * hi);

  float carry = 0.0f;  // chunk-local cum at start of the next (later) tile
  for (int tile = TILES - 1; tile >= 0; --tile) {
    const int tb = tile * 16;
    v8f acc = {};
#pragma unroll
    for (int kk = 0; kk < 4; ++kk) {
      const int s = tb + 4 * kk + 2 * hi;   // K index = time s within tile
      v2f bfr;
      bfr.x = sm[ln * LSTR + s];
      bfr.y = sm[ln * LSTR + s + 1];
      // D = A x B + C  (chained K accumulation, f32)
      acc = __builtin_amdgcn_wmma_f32_16x16x4_f32(false, afrag[kk], false, bfr,
                                                  (short)0, acc, false, false);
    }
    // Fold in carry from later tiles and store local results.
    float full0 = 0.0f;
    const int base = (b0 + ln) * Tn + t0 + tb + 8 * hi;
#pragma unroll
    for (int v = 0; v < 8; ++v) {
      const float f = acc[v] + gf[v] * carry;
      if (v == 0) full0 = f;
      cum[base + v] = f;     // scattered 4B, absorbed by 192MB L2
    }
    // New carry = value at time tb (VGPR0, hi==0 lanes), broadcast per row.
    carry = __shfl(full0, ln, 32);
  }
  // Chunk-start local prefix (needed for inter-chunk combine).
  if (hi == 0) Sbuf[(b0 + ln) * NCH + chunk] = carry;
}

// ---------------------------------------------------------------------------
// Kernel 2: per-row reverse scan over 32 chunk aggregates.
// In-place: Sbuf[b][j] becomes carryIn for chunk j (= CB[j+1]).
// ---------------------------------------------------------------------------
__global__ void k_chunk_scan(float* __restrict__ Sbuf)
{
  const int b = blockIdx.x * blockDim.x + threadIdx.x;
  if (b >= Bn) return;
  float gC = 1.0f;
  for (int i = 0; i < CHUNK; ++i) gC *= GAMMA;   // gamma^256
  float carry = 0.0f;
  for (int j = NCH - 1; j >= 0; --j) {
    const float s = Sbuf[b * NCH + j];
    Sbuf[b * NCH + j] = carry;       // carryIn[j] = CB[j+1]
    carry = s + gC * carry;          // CB[j]
  }
}

// ---------------------------------------------------------------------------
// Kernel 3: fully-coalesced float4 finalize.
// cum += gamma^(CHUNK - t_local) * carryIn ; obj = clip(w*cum) * log_probs.
// ---------------------------------------------------------------------------
__global__ void k_finalize(const float* __restrict__ log_probs,
                           const float* __restrict__ weight,
                           const float* __restrict__ Sbuf,
                           float* __restrict__ obj,
                           float* __restrict__ cum)
{
  const int tid = blockIdx.x * blockDim.x + threadIdx.x;   // one float4 each
  const int i4  = tid * 4;
  const int b   = i4 / Tn;
  const int t   = i4 % Tn;
  const int j   = t / CHUNK;
  const int tl  = t % CHUNK;      // multiple of 4; group never crosses chunks

  const float ci  = Sbuf[b * NCH + j];
  const float L2G = -0.0144995696951151f;        // log2(0.99)
  float f = exp2f((float)(CHUNK - tl) * L2G);    // gamma^(CHUNK - tl)
  const float inv = 1.0f / GAMMA;

  float4 c4 = ((const float4*)cum)[tid];
  float4 w4 = ((const float4*)weight)[tid];
  float4 p4 = ((const float4*)log_probs)[tid];

  c4.x += f * ci; f *= inv;
  c4.y += f * ci; f *= inv;
  c4.z += f * ci; f *= inv;
  c4.w += f * ci;

  float4 o4;
  o4.x = clampv(w4.x * c4.x) * p4.x;
  o4.y = clampv(w4.y * c4.y) * p4.y;
  o4.z = clampv(w4.z * c4.z) * p4.z;
  o4.w = clampv(w4.w * c4.w) * p4.w;

  ((float4*)cum)[tid] = c4;
  ((float4*)obj)[tid] = o4;
}

// ---------------------------------------------------------------------------
extern "C" void kernel_launch(void* const* d_in, const int* in_sizes, int n_in,
                              void* d_out, int out_size, void* d_ws, size_t ws_size,
                              hipStream_t stream)
{
  const float* log_probs = (const float*)d_in[0];   // [B,T]
  const float* logits    = (const float*)d_in[1];   // [B,T,1]
  const float* weight    = (const float*)d_in[2];   // [B,T]
  // d_in[3] = baselines: unused by the reference.

  float* obj = (float*)d_out;                       // [B,T]
  float* cum = obj + (size_t)Bn * Tn;               // [B,T] (also scan staging)
  float* Sb  = (float*)d_ws;                        // [B, NCH] = 128 KB

  // 1) local scans: (B/16)*NCH = 2048 waves, WPB waves per block
  const int units  = (Bn / 16) * NCH;
  const int blocks = units / WPB;
  k_local_scan<<<blocks, WPB * 32, 0, stream>>>(logits, weight, cum, Sb);

  // 2) inter-chunk carry scan: one thread per row
  k_chunk_scan<<<(Bn + 255) / 256, 256, 0, stream>>>(Sb);

  // 3) finalize: one float4 per thread
  const int n4 = (Bn * Tn) / 4;
  k_finalize<<<n4 / 256, 256, 0, stream>>>(log_probs, weight, Sb, obj, cum);
}